// CopyGenerator_38706245271880
// MI455X (gfx1250) — compile-verified
//
#include <hip/hip_runtime.h>
#include <hip/hip_bf16.h>
#include <math.h>

typedef __attribute__((ext_vector_type(2))) float v2f;
typedef __attribute__((ext_vector_type(8))) float v8f;

#define B_  8
#define T_  256
#define S_  400
#define H_  512
#define V_  32000
#define ROWS (B_*T_)   // 2048

// ---------------------------------------------------------------------------
// 1) p_gen = sigmoid(state_input (2048x1024) . w_pgen (1024) + b_pgen)
//    one wave32 per row, 8 rows per 256-thread block
// ---------------------------------------------------------------------------
__global__ void pgen_kernel(const float* __restrict__ state_input,
                            const float* __restrict__ w_pgen,
                            const float* __restrict__ b_pgen,
                            float* __restrict__ pgen) {
    const int wave = threadIdx.x >> 5;
    const int lane = threadIdx.x & 31;
    const int row  = blockIdx.x * 8 + wave;
    if (row >= ROWS) return;
    const float* si = state_input + (size_t)row * (2 * H_);
    float sum = 0.0f;
    #pragma unroll 8
    for (int i = lane; i < 2 * H_; i += 32)
        sum += si[i] * w_pgen[i];
    #pragma unroll
    for (int off = 16; off > 0; off >>= 1)
        sum += __shfl_down(sum, off, 32);
    if (lane == 0) {
        float z = sum + b_pgen[0];
        pgen[row] = 1.0f / (1.0f + __expf(-z));
    }
}

// ---------------------------------------------------------------------------
// 2/3) C(row-major MxN) = A(MxK=512) @ B(KxN) + bias(N), fp32 WMMA 16x16x4.
//    N is a template constant -> row strides fold into immediate offsets /
//    GVS (saddr + 32-bit voffset) addressing, no 64-bit VGPR address chains.
//    Block: 256 threads = 8 waves. Each wave owns TWO independent 16x16
//    tiles (16x32) so two WMMA dependency chains interleave in the matrix
//    pipe. Block covers 16 rows x 256 cols. grid.x = N/256, grid.y = M/16.
// ---------------------------------------------------------------------------
#define GEMM_K 512
#define LDSP  513   // pad: stride 513 % 64 != 0 -> conflict-free column reads

template<int N>
__global__ void gemm_wmma_f32(const float* __restrict__ A,
                              const float* __restrict__ Bm,
                              const float* __restrict__ bias,
                              float* __restrict__ C) {
    __shared__ float As[16 * LDSP];

    const int tid  = threadIdx.x;
    const int wave = tid >> 5;
    const int lane = tid & 31;
    const int half = lane >> 4;     // 0: lanes 0-15, 1: lanes 16-31
    const int lm   = lane & 15;

    const int m0 = blockIdx.y * 16;
    const int n0 = blockIdx.x * 256 + wave * 32;

    // cooperative load of the 16x512 A tile (32 floats per thread, coalesced)
    const float* Ab = A + (size_t)m0 * GEMM_K;
    #pragma unroll
    for (int i = 0; i < (16 * GEMM_K) / 256; ++i) {
        int e = tid + i * 256;
        int m = e >> 9;          // /512
        int k = e & (GEMM_K - 1);
        As[m * LDSP + k] = Ab[m * GEMM_K + k];
    }
    __syncthreads();

    v8f acc0 = {};
    v8f acc1 = {};
    const int arow = lm * LDSP;
    // per-lane B index: (K = 2*half) row, column n0+lm  (32-bit, GVS-friendly)
    int boff = (2 * half) * N + n0 + lm;

    #pragma unroll 4
    for (int kk = 0; kk < GEMM_K; kk += 4) {
        v2f a;
        a.x = As[arow + kk + 2 * half];          // A[m=lm][K=2*half+0]
        a.y = As[arow + kk + 2 * half + 1];      // A[m=lm][K=2*half+1]
        v2f b0, b1;
        b0.x = Bm[boff];                         // tile0: B[K+0][n]
        b0.y = Bm[boff + N];                     // tile0: B[K+1][n]
        b1.x = Bm[boff + 16];                    // tile1: B[K+0][n+16]
        b1.y = Bm[boff + N + 16];                // tile1: B[K+1][n+16]
        acc0 = __builtin_amdgcn_wmma_f32_16x16x4_f32(
            false, a, false, b0, (short)0, acc0, false, false);
        acc1 = __builtin_amdgcn_wmma_f32_16x16x4_f32(
            false, a, false, b1, (short)0, acc1, false, false);
        boff += 4 * N;
    }

    // D layout: VGPR r -> (M = r + 8*half, N = lm)
    const int col = n0 + lm;
    const float bv0 = bias[col];
    const float bv1 = bias[col + 16];
    const int co = (m0 + 8 * half) * N + col;    // 32-bit base index
    #pragma unroll
    for (int r = 0; r < 8; ++r) {
        C[co + r * N]      = acc0[r] + bv0;      // r*N folds to immediate
        C[co + r * N + 16] = acc1[r] + bv1;
    }
}

// ---------------------------------------------------------------------------
// 4) per-row: out[v] = pgen[row] * softmax(out)[v]
//    one 256-thread block per row; 3 passes (max, sumexp, scale); the 128KB
//    row stays resident in L2 between passes.
// ---------------------------------------------------------------------------
__global__ void softmax_scale_kernel(float* __restrict__ out,
                                     const float* __restrict__ pgen) {
    __shared__ float red[256];
    const int tid = threadIdx.x;
    const int row = blockIdx.x;
    float* x = out + (size_t)row * V_;

    // pass 1: row max
    float lmax = -3.402823466e+38f;
    for (int i = tid; i < V_; i += 256) lmax = fmaxf(lmax, x[i]);
    red[tid] = lmax; __syncthreads();
    #pragma unroll
    for (int off = 128; off > 0; off >>= 1) {
        if (tid < off) red[tid] = fmaxf(red[tid], red[tid + off]);
        __syncthreads();
    }
    const float rmax = red[0]; __syncthreads();

    // pass 2: sum of exp
    float lsum = 0.0f;
    for (int i = tid; i < V_; i += 256) lsum += __expf(x[i] - rmax);
    red[tid] = lsum; __syncthreads();
    #pragma unroll
    for (int off = 128; off > 0; off >>= 1) {
        if (tid < off) red[tid] += red[tid + off];
        __syncthreads();
    }
    const float rsum = red[0];

    // pass 3: scale by p_gen / sum
    const float scale = pgen[row] / rsum;
    for (int i = tid; i < V_; i += 256) x[i] = __expf(x[i] - rmax) * scale;
}

// ---------------------------------------------------------------------------
// 5) scatter: out[row, enc[b,s]] += (1 - pgen[row]) * attn[row, s]
// ---------------------------------------------------------------------------
__global__ void scatter_kernel(float* __restrict__ out,
                               const float* __restrict__ attn,
                               const int* __restrict__ enc,
                               const float* __restrict__ pgen) {
    const int gid = blockIdx.x * blockDim.x + threadIdx.x;
    if (gid >= B_ * T_ * S_) return;
    const int s   = gid % S_;
    const int row = gid / S_;          // row = b*T + t
    const int b   = row / T_;
    const float w = 1.0f - pgen[row];
    const int idx = enc[b * S_ + s];
    atomicAdd(&out[(size_t)row * V_ + idx], w * attn[(size_t)row * S_ + s]);
}

// ---------------------------------------------------------------------------
// 6) out = log(out + 1e-12), float4-vectorized
// ---------------------------------------------------------------------------
__global__ void log_kernel(float4* __restrict__ out, int n4) {
    const int gid = blockIdx.x * blockDim.x + threadIdx.x;
    if (gid >= n4) return;
    float4 v = out[gid];
    v.x = __logf(v.x + 1e-12f);
    v.y = __logf(v.y + 1e-12f);
    v.z = __logf(v.z + 1e-12f);
    v.w = __logf(v.w + 1e-12f);
    out[gid] = v;
}

// ---------------------------------------------------------------------------
extern "C" void kernel_launch(void* const* d_in, const int* in_sizes, int n_in,
                              void* d_out, int out_size, void* d_ws, size_t ws_size,
                              hipStream_t stream) {
    const float* s_output    = (const float*)d_in[0];  // (2048, 512)
    const float* state_input = (const float*)d_in[1];  // (2048, 1024)
    const float* attn        = (const float*)d_in[2];  // (2048, 400)
    const int*   enc         = (const int*)  d_in[3];  // (8, 400)
    const float* w_pgen      = (const float*)d_in[4];  // (1024, 1)
    const float* b_pgen      = (const float*)d_in[5];  // (1,)
    const float* w1          = (const float*)d_in[6];  // (512, 512)
    const float* b1          = (const float*)d_in[7];  // (512,)
    const float* w2          = (const float*)d_in[8];  // (512, 32000)
    const float* b2          = (const float*)d_in[9];  // (32000,)

    float* out  = (float*)d_out;                       // (2048, 32000)
    float* pgen = (float*)d_ws;                        // 2048 floats
    float* X    = pgen + ROWS;                         // (2048, 512) hidden

    // 1) p_gen
    pgen_kernel<<<ROWS / 8, 256, 0, stream>>>(state_input, w_pgen, b_pgen, pgen);

    // 2) X = s_output @ w1 + b1   (M=2048, N=512, K=512)
    gemm_wmma_f32<H_><<<dim3(H_ / 256, ROWS / 16), 256, 0, stream>>>(
        s_output, w1, b1, X);

    // 3) logits = X @ w2 + b2 -> d_out   (M=2048, N=32000, K=512)
    gemm_wmma_f32<V_><<<dim3(V_ / 256, ROWS / 16), 256, 0, stream>>>(
        X, w2, b2, out);

    // 4) out = p_gen * softmax(out) per row
    softmax_scale_kernel<<<ROWS, 256, 0, stream>>>(out, pgen);

    // 5) scatter-add copy distribution
    scatter_kernel<<<(B_ * T_ * S_ + 255) / 256, 256, 0, stream>>>(
        out, attn, enc, pgen);

    // 6) log(out + 1e-12)
    const int n4 = (ROWS * V_) / 4;
    log_kernel<<<(n4 + 255) / 256, 256, 0, stream>>>((float4*)out, n4);
}